// FactorizationMachine_26156350832970
// MI455X (gfx1250) — compile-verified
//
#include <hip/hip_runtime.h>
#include <stdint.h>

#ifndef __has_builtin
#define __has_builtin(x) 0
#endif

#if __has_builtin(__builtin_amdgcn_wmma_f32_16x16x4_f32)
#define USE_WMMA 1
#else
#define USE_WMMA 0
#endif

#if __has_builtin(__builtin_amdgcn_global_load_async_to_lds_b128)
#define USE_ASYNC 1
#else
#define USE_ASYNC 0
#endif

#if __has_builtin(__builtin_amdgcn_s_wait_asynccnt)
#define WAIT_ASYNC() __builtin_amdgcn_s_wait_asynccnt(0)
#else
#define WAIT_ASYNC() asm volatile("s_wait_asynccnt 0x0" ::: "memory")
#endif

namespace {
constexpr int kF = 26;
constexpr int kCard = 100;
constexpr int kE = 16;
constexpr int kRows = kF * kCard;             // 2600 gathered rows per element
constexpr int kSparseCols = kF * (kCard + 1); // 2626
constexpr int kDense = 13;
constexpr int kElemsPerWG = 8;                // 8 waves, 1 element per wave
constexpr int kBlock = 256;
}

typedef float v8f __attribute__((ext_vector_type(8)));
typedef float v2f __attribute__((ext_vector_type(2)));

// Exact pointee type the async-LDS builtin expects: GCC vector of 4 ints
typedef int gcc_v4i __attribute__((vector_size(16)));
typedef __attribute__((address_space(1))) gcc_v4i* as1_v4i_p;
typedef __attribute__((address_space(3))) gcc_v4i* as3_v4i_p;

__global__ __launch_bounds__(kBlock, 1)
void fm_fused(const int* __restrict__ sparse,
              const float* __restrict__ dense,
              const float* __restrict__ lin_tab,
              const float* __restrict__ emb_tab,
              const float* __restrict__ lw,
              const float* __restrict__ lb,
              float* __restrict__ out,
              int batch)
{
    // 166,400 + 10,400 + 41,600 = 218,400 B of the 320 KB WGP LDS
    __shared__ __align__(16) float          s_emb[kRows * kE];
    __shared__ __align__(16) float          s_linw[kRows];
    __shared__ __align__(16) unsigned short s_idx[kElemsPerWG * kRows];

    const int tid = threadIdx.x;
    const int b0  = blockIdx.x * kElemsPerWG;

    // ---- Stage the full embedding table into LDS (float4 granularity) ----
    {
        const float4* g4 = reinterpret_cast<const float4*>(emb_tab);
#if USE_ASYNC
        for (int t = tid; t < kRows * kE / 4; t += kBlock) {
            as1_v4i_p gp = (as1_v4i_p)(uintptr_t)(g4 + t);
            as3_v4i_p lp = (as3_v4i_p)(uint32_t)(uintptr_t)(&s_emb[t * 4]);
            __builtin_amdgcn_global_load_async_to_lds_b128(gp, lp, 0, 0);
        }
#else
        float4* s4 = reinterpret_cast<float4*>(s_emb);
        for (int t = tid; t < kRows * kE / 4; t += kBlock) s4[t] = g4[t];
#endif
    }

    // ---- Stage linear table with linear_w[f] pre-folded in ----
    for (int t = tid; t < kRows; t += kBlock) {
        int f = t / kCard;
        s_linw[t] = lin_tab[t] * lw[f];
    }

    // ---- Stage compacted row-ids (f*100 + idx) as u16, one set per wave ----
    for (int t = tid; t < kElemsPerWG * kRows; t += kBlock) {
        int e = t / kRows;
        int r = t - e * kRows;          // r = f*100 + c
        int f = r / kCard;
        int b = b0 + e;
        unsigned short rowid = 0;
        if (b < batch) {
            // sparse column = f*101 + c = r + f
            int idx = sparse[(size_t)b * kSparseCols + r + f];
            rowid = (unsigned short)(f * kCard + idx);
        }
        s_idx[t] = rowid;
    }

#if USE_ASYNC
    WAIT_ASYNC();
#endif
    __syncthreads();

    // ---- One batch element per wave32 ----
    const int wave = tid >> 5;
    const int lane = tid & 31;
    const int b = b0 + wave;
    if (b >= batch) return;

    const unsigned short* idxp = s_idx + wave * kRows;

    // Linear term: dense part + bias seeded per-lane, then gather-sum
    float lin = 0.f;
    if (lane < kDense)       lin = dense[(size_t)b * kDense + lane] * lw[kF + lane];
    else if (lane == kDense) lin = lb[0];
    for (int r = lane; r < kRows; r += 32) lin += s_linw[idxp[r]];

    const int e16  = lane & 15;
    const int half = lane >> 4;

    float p, diag;
#if USE_WMMA
    // Gram matrix G = w^T w via V_WMMA_F32_16X16X4_F32: the f32 A-transposed
    // layout equals the B layout here, so one float2 load feeds both operands.
    // A second WMMA with A = ones accumulates s = 1^T w into every row of accS.
    v8f accG0 = {}, accG1 = {}, accS0 = {}, accS1 = {};
    const v2f ones = {1.f, 1.f};
    for (int k0 = 0; k0 < kRows; k0 += 8) {
        unsigned long long q0 = *reinterpret_cast<const unsigned long long*>(idxp + k0);
        unsigned long long q1 = *reinterpret_cast<const unsigned long long*>(idxp + k0 + 4);
        unsigned p0 = half ? (unsigned)(q0 >> 32) : (unsigned)q0;
        unsigned p1 = half ? (unsigned)(q1 >> 32) : (unsigned)q1;
        v2f a0, a1;
        a0.x = s_emb[(p0 & 0xffffu) * kE + e16];
        a0.y = s_emb[(p0 >> 16)     * kE + e16];
        a1.x = s_emb[(p1 & 0xffffu) * kE + e16];
        a1.y = s_emb[(p1 >> 16)     * kE + e16];
        accG0 = __builtin_amdgcn_wmma_f32_16x16x4_f32(false, a0,   false, a0, (short)0, accG0, false, false);
        accS0 = __builtin_amdgcn_wmma_f32_16x16x4_f32(false, ones, false, a0, (short)0, accS0, false, false);
        accG1 = __builtin_amdgcn_wmma_f32_16x16x4_f32(false, a1,   false, a1, (short)0, accG1, false, false);
        accS1 = __builtin_amdgcn_wmma_f32_16x16x4_f32(false, ones, false, a1, (short)0, accS1, false, false);
    }
    v8f accG = accG0 + accG1;
    v8f accS = accS0 + accS1;

    // accS VGPR0 holds s[e16] in every lane (rows M=0 and M=8 of the s-matrix)
    p = accS[0] * accS[0];  // wave-sum counts each e twice -> *0.5 later

    // Diagonal of C/D layout: lane v (VGPR v) has G[v][v]; lane v+24 has G[v+8][v+8]
    diag = 0.f;
#pragma unroll
    for (int v = 0; v < 8; ++v) {
        if (lane == v)      diag = accG[v];
        if (lane == v + 24) diag = accG[v];
    }
#else
    // VALU fallback (same math)
    float sacc = 0.f, soacc = 0.f;
    for (int k0 = 0; k0 < kRows; k0 += 4) {
        unsigned long long q0 = *reinterpret_cast<const unsigned long long*>(idxp + k0);
        unsigned p0 = half ? (unsigned)(q0 >> 32) : (unsigned)q0;
        float x = s_emb[(p0 & 0xffffu) * kE + e16];
        float y = s_emb[(p0 >> 16)     * kE + e16];
        sacc  += x + y;
        soacc += x * x + y * y;
    }
    sacc += __shfl_xor(sacc, 16, 32);  // combine half-waves -> s[e16] per lane
    p = sacc * sacc;                   // duplicated across halves -> *0.5 later
    diag = soacc;                      // wave-sum = trace
#endif

    // Wave reduction of (sum s^2 * 2, trace, linear)
#pragma unroll
    for (int off = 16; off > 0; off >>= 1) {
        p    += __shfl_xor(p,    off, 32);
        diag += __shfl_xor(diag, off, 32);
        lin  += __shfl_xor(lin,  off, 32);
    }

    if (lane == 0) {
        float sum_s2 = 0.5f * p;
        out[b] = lin + 0.5f * (sum_s2 - diag);
    }
}

extern "C" void kernel_launch(void* const* d_in, const int* in_sizes, int n_in,
                              void* d_out, int out_size, void* d_ws, size_t ws_size,
                              hipStream_t stream) {
    (void)in_sizes; (void)n_in; (void)d_ws; (void)ws_size;
    const int*   sparse  = (const int*)d_in[0];
    const float* dense   = (const float*)d_in[1];
    const float* lin_tab = (const float*)d_in[2];
    const float* emb_tab = (const float*)d_in[3];
    const float* lw      = (const float*)d_in[4];
    const float* lb      = (const float*)d_in[5];
    float* out = (float*)d_out;

    int batch = out_size;  // (B,1) f32 output, flat = B
    int grid  = (batch + kElemsPerWG - 1) / kElemsPerWG;
    fm_fused<<<grid, kBlock, 0, stream>>>(sparse, dense, lin_tab, emb_tab, lw, lb, out, batch);
}